// AdaptivePrecisionKVCache_88562225643883
// MI455X (gfx1250) — compile-verified
//
#include <hip/hip_runtime.h>
#include <stdint.h>

// AdaptivePrecisionKVCache for MI455X (gfx1250).
// Pass 1: global masked min/max reduction: b128 loads (fills 192MB L2),
//         wave32 shuffle tree -> LDS cross-wave reduce -> ONE atomic set per
//         block (order-preserving-uint encodings so u32 atomics do float
//         min/max; avoids same-address atomic serialization at L2).
// Pass 2: elementwise quant/dequant streamed through LDS with
//         global_load_async_to_lds_b128 double buffering (ASYNCcnt paced),
//         non-temporal b128 output stores so the input stays L2-resident.

#define KV_THRESH 0.01f

typedef float v4f __attribute__((ext_vector_type(4)));

// ---- order-preserving float <-> uint so integer atomics do float min/max ----
__device__ __forceinline__ unsigned kv_f2o(float f) {
    unsigned u = __float_as_uint(f);
    return (u & 0x80000000u) ? ~u : (u | 0x80000000u);
}
__device__ __forceinline__ float kv_o2f(unsigned e) {
    unsigned u = (e & 0x80000000u) ? (e & 0x7fffffffu) : ~e;
    return __uint_as_float(u);
}

// ws layout (uint32): [0]=min_small [1]=max_small [2]=min_large [3]=max_large
__global__ void kv_init_ws(unsigned* ws) {
    int t = threadIdx.x;
    if (t < 4) ws[t] = (t & 1) ? kv_f2o(-__builtin_inff()) : kv_f2o(__builtin_inff());
}

__global__ __launch_bounds__(256) void kv_minmax(const float* __restrict__ in,
                                                 unsigned* __restrict__ ws,
                                                 int nvec, int n) {
    __shared__ float red[4][8];                       // [value][wave]; 8 waves/block

    float smin = __builtin_inff(), smax = -__builtin_inff();
    float lmin = __builtin_inff(), lmax = -__builtin_inff();
    const int stride = gridDim.x * blockDim.x;
    const v4f* in4 = (const v4f*)in;

    for (int i = blockIdx.x * blockDim.x + threadIdx.x; i < nvec; i += stride) {
        v4f v = in4[i];                               // global_load_b128, RT: fill L2
#pragma unroll
        for (int c = 0; c < 4; ++c) {
            float x = v[c];
            if (__builtin_fabsf(x) > KV_THRESH) {
                lmin = fminf(lmin, x); lmax = fmaxf(lmax, x);
            } else {
                smin = fminf(smin, x); smax = fmaxf(smax, x);
            }
        }
    }
    // scalar tail (n not a multiple of 4 — not hit for this shape)
    for (int j = 4 * nvec + blockIdx.x * blockDim.x + threadIdx.x; j < n; j += stride) {
        float x = in[j];
        if (__builtin_fabsf(x) > KV_THRESH) {
            lmin = fminf(lmin, x); lmax = fmaxf(lmax, x);
        } else {
            smin = fminf(smin, x); smax = fmaxf(smax, x);
        }
    }

    // wave32 butterfly reduction
#pragma unroll
    for (int off = 16; off; off >>= 1) {
        smin = fminf(smin, __shfl_xor(smin, off, 32));
        smax = fmaxf(smax, __shfl_xor(smax, off, 32));
        lmin = fminf(lmin, __shfl_xor(lmin, off, 32));
        lmax = fmaxf(lmax, __shfl_xor(lmax, off, 32));
    }

    // cross-wave reduce in LDS -> one atomic set per block
    const int wave = threadIdx.x >> 5;
    if ((threadIdx.x & 31) == 0) {
        red[0][wave] = smin; red[1][wave] = smax;
        red[2][wave] = lmin; red[3][wave] = lmax;
    }
    __syncthreads();
    if (threadIdx.x < 32) {
        int w = threadIdx.x & 7;                      // lanes replicate over 8 waves
        float a = red[0][w], b = red[1][w], c = red[2][w], d = red[3][w];
#pragma unroll
        for (int off = 4; off; off >>= 1) {
            a = fminf(a, __shfl_xor(a, off, 32));
            b = fmaxf(b, __shfl_xor(b, off, 32));
            c = fminf(c, __shfl_xor(c, off, 32));
            d = fmaxf(d, __shfl_xor(d, off, 32));
        }
        if (threadIdx.x == 0) {
            atomicMin(ws + 0, kv_f2o(a));
            atomicMax(ws + 1, kv_f2o(b));
            atomicMin(ws + 2, kv_f2o(c));
            atomicMax(ws + 3, kv_f2o(d));
        }
    }
}

__device__ __forceinline__ float kv_dq(float x,
                                       float smin, float sden, float sstp, bool sok,
                                       float lmin, float lden, float lstp, bool lok) {
    bool  big  = __builtin_fabsf(x) > KV_THRESH;
    float bmin = big ? lmin : smin;
    float den  = big ? lden : sden;
    float stp  = big ? lstp : sstp;                   // den/levels, hoisted division
    float lv   = big ? 255.0f : 15.0f;
    bool  ok   = big ? lok : sok;
    // rounding-critical path kept as exact IEEE div (matches jnp bit behavior);
    // dequant step uses hoisted den/levels (<=2ulp-of-den perturbation).
    float q = rintf((x - bmin) / den * lv);
    float d = q * stp + bmin;
    return ok ? d : x;
}

// per-lane async global->LDS b128 copy (CDNA5 async memory path)
__device__ __forceinline__ void kv_async_b128(const float* g, v4f* l) {
    uint32_t loff = (uint32_t)(uintptr_t)l;           // LDS byte offset (low 32b of flat)
    uint64_t gaddr = (uint64_t)(uintptr_t)g;
    asm volatile("global_load_async_to_lds_b128 %0, %1, off"
                 :: "v"(loff), "v"(gaddr) : "memory");
}

__global__ __launch_bounds__(256) void kv_dequant(const float* __restrict__ in,
                                                  const unsigned* __restrict__ ws,
                                                  float* __restrict__ out,
                                                  int nvec, int n) {
    __shared__ v4f stage[2][256];                     // 8 KiB, double buffer

    float smin = kv_o2f(ws[0]), smax = kv_o2f(ws[1]);
    float lmin = kv_o2f(ws[2]), lmax = kv_o2f(ws[3]);
    bool  sok = smax > smin,  lok = lmax > lmin;      // has_any && (max != min)
    float sden = smax - smin, lden = lmax - lmin;
    float sstp = sden / 15.0f, lstp = lden / 255.0f;  // uniform, one div each

    const int tid = threadIdx.x;
    const int stride = gridDim.x * blockDim.x;
    int i = blockIdx.x * blockDim.x + tid;
    int buf = 0;

    if (i < nvec) kv_async_b128(in + 4 * (size_t)i, &stage[0][tid]);  // prologue

    for (; i < nvec; i += stride) {
        int nxt = i + stride;
        if (nxt < nvec) {
            kv_async_b128(in + 4 * (size_t)nxt, &stage[buf ^ 1][tid]); // prefetch
            asm volatile("s_wait_asynccnt 0x1" ::: "memory");          // oldest done
        } else {
            asm volatile("s_wait_asynccnt 0x0" ::: "memory");
        }
        v4f v = stage[buf][tid];                       // ds_load_b128
        v4f r;
#pragma unroll
        for (int c = 0; c < 4; ++c)
            r[c] = kv_dq(v[c], smin, sden, sstp, sok, lmin, lden, lstp, lok);
        __builtin_nontemporal_store(r, (v4f*)out + i); // NT b128: keep input in L2
        asm volatile("s_wait_dscnt 0x0" ::: "memory"); // LDS read done before reuse
        buf ^= 1;
    }

    // scalar tail (not hit for this shape)
    for (int j = 4 * nvec + blockIdx.x * blockDim.x + tid; j < n; j += stride)
        out[j] = kv_dq(in[j], smin, sden, sstp, sok, lmin, lden, lstp, lok);
}

extern "C" void kernel_launch(void* const* d_in, const int* in_sizes, int n_in,
                              void* d_out, int out_size, void* d_ws, size_t ws_size,
                              hipStream_t stream) {
    const float* in  = (const float*)d_in[0];
    float*       out = (float*)d_out;
    unsigned*    ws  = (unsigned*)d_ws;

    int n    = in_sizes[0];      // 2*16*8192*128 = 33,554,432
    int nvec = n >> 2;           // 8,388,608 float4s

    kv_init_ws<<<1, 32, 0, stream>>>(ws);

    int rblocks = (nvec + 255) / 256;
    if (rblocks > 2048) rblocks = 2048;   // 16 b128 iters/thread; 2K atomics/addr
    if (rblocks < 1) rblocks = 1;
    kv_minmax<<<rblocks, 256, 0, stream>>>(in, ws, nvec, n);

    int dblocks = (nvec + 255) / 256;
    if (dblocks > 4096) dblocks = 4096;   // 8 pipelined iters per thread
    if (dblocks < 1) dblocks = 1;
    kv_dequant<<<dblocks, 256, 0, stream>>>(in, ws, out, nvec, n);
}